// TradingLoss_83356725281290
// MI455X (gfx1250) — compile-verified
//
#include <hip/hip_runtime.h>
#include <stdint.h>

// ---------------- Problem constants ----------------
#define B_ROWS 8192
#define N_COLS 1024

// ---------------- Tiling ----------------
#define BROWS   128                  // rows per block == threads per block (4 wave32s)
#define TCOLS   32                   // columns per tile
#define NT      (N_COLS / TCOLS)     // 32 tiles
#define LSTRIDE 36                   // DWORDs per LDS row: 32 data + 4 pad (16B aligned, avoids bank collisions)
#define ARR_STRIDE (BROWS * LSTRIDE)     // DWORDs per array (pred or price) per buffer
#define BUF_STRIDE (2 * ARR_STRIDE)      // DWORDs per buffer (pred + price)

// ---------------- CDNA5 async global->LDS (ASYNCcnt) ----------------
__device__ __forceinline__ void async_copy_b128(uint32_t lds_byte_off, uint64_t gaddr) {
  // Per-lane DMA: copies 16 bytes from the global address in gaddr to the LDS address in lds_byte_off, tracked by ASYNCcnt.
  asm volatile("global_load_async_to_lds_b128 %0, %1, off"
               :: "v"(lds_byte_off), "v"(gaddr)
               : "memory");
}
__device__ __forceinline__ void wait_async_le16() {
  asm volatile("s_wait_asynccnt 16" ::: "memory");
}
__device__ __forceinline__ void wait_async_0() {
  asm volatile("s_wait_asynccnt 0" ::: "memory");
}

// Issue one 128x32 tile (pred + price) into LDS buffer `buf`: 16 async B128 ops/thread.
__device__ __forceinline__ void issue_tile(const float* __restrict__ pred,
                                           const float* __restrict__ price,
                                           int rowBase, int t,
                                           uint32_t lds_base, int buf, int tid) {
  const uint32_t bo = lds_base + (uint32_t)buf * (BUF_STRIDE * 4u);
  const int colBase = t * TCOLS;
#pragma unroll
  for (int i = 0; i < 8; ++i) {
    int Q   = i * BROWS + tid;     // quad index within tile, 0..1023
    int row = Q >> 3;              // 8 quads (of 4 floats) per 32-col row
    int q   = Q & 7;
    uint32_t loff = (uint32_t)row * (LSTRIDE * 4u) + (uint32_t)q * 16u;
    size_t gelem  = (size_t)(rowBase + row) * N_COLS + (size_t)(colBase + q * 4);
    async_copy_b128(bo + loff,                   (uint64_t)(uintptr_t)(pred  + gelem));
    async_copy_b128(bo + ARR_STRIDE * 4u + loff, (uint64_t)(uintptr_t)(price + gelem));
  }
}

// ---------------- Kernel A: per-row trading scan + block partials ----------------
__global__ __launch_bounds__(BROWS) void trade_scan_kernel(const float* __restrict__ pred,
                                                           const float* __restrict__ price,
                                                           float* __restrict__ partials) {
  __shared__ float sh[2 * BUF_STRIDE];   // double-buffered tiles: ~72 KB
  __shared__ float red[3][BROWS];

  const int tid     = threadIdx.x;
  const int rowBase = blockIdx.x * BROWS;
  const uint32_t lds_base = (uint32_t)(uintptr_t)&sh[0];

  // prefetch tile 0
  issue_tile(pred, price, rowBase, 0, lds_base, 0, tid);

  float cash   = 10000.0f;   // INITIAL_CASH
  int   pos    = 0;
  float absacc = 0.0f;
  float lastp  = 0.0f;

  for (int t = 0; t < NT; ++t) {
    const int buf = t & 1;
    if (t + 1 < NT) {
      issue_tile(pred, price, rowBase, t + 1, lds_base, buf ^ 1, tid);
      wait_async_le16();               // tile t complete; tile t+1 still in flight
    } else {
      wait_async_0();
    }
    __syncthreads();

    const float* prow = &sh[(size_t)buf * BUF_STRIDE + (size_t)tid * LSTRIDE];
    const float* crow = prow + ARR_STRIDE;

#pragma unroll 4
    for (int j = 0; j < TCOLS; ++j) {
      float pv = prow[j];              // trading prediction
      float p  = crow[j];              // price (strictly > 0)
      absacc += fabsf(pv);

      // action: round half-to-even, clamp to [-10, 10]
      float af = fminf(10.0f, fmaxf(-10.0f, rintf(pv)));
      int a = (int)af;

      // buy branch (0 when a <= 0)
      int ab = max(0, min(a, 10 - pos));
      float req = (float)ab * p;
      int buy = ab;
      if (cash < req) {
        int afford = (int)floorf(cash / p);
        buy = max(0, min(afford, ab));
      }
      // sell branch (0 when a >= 0)
      int sell = max(0, min(-a, pos + 10));

      cash += (float)(sell - buy) * p;
      pos  += buy - sell;
      lastp = p;
    }
    __syncthreads();                   // protect buf before tile t+2 overwrites it
  }

  const float value = cash + (float)pos * lastp;
  const float r = (value - 10000.0f) * (1.0f / 10000.0f);

  red[0][tid] = r;
  red[1][tid] = r * r;
  red[2][tid] = absacc;
  __syncthreads();
#pragma unroll
  for (int s = BROWS >> 1; s > 0; s >>= 1) {
    if (tid < s) {
      red[0][tid] += red[0][tid + s];
      red[1][tid] += red[1][tid + s];
      red[2][tid] += red[2][tid + s];
    }
    __syncthreads();
  }
  if (tid == 0) {
    partials[blockIdx.x * 3 + 0] = red[0][0];
    partials[blockIdx.x * 3 + 1] = red[1][0];
    partials[blockIdx.x * 3 + 2] = red[2][0];
  }
}

// ---------------- Kernel B: single-wave32 finalize ----------------
__global__ void finalize_kernel(const float* __restrict__ partials,
                                float* __restrict__ out, int nblk) {
  const int lane = threadIdx.x;
  float sR = 0.0f, sR2 = 0.0f, sA = 0.0f;
  for (int i = lane; i < nblk; i += 32) {
    sR  += partials[i * 3 + 0];
    sR2 += partials[i * 3 + 1];
    sA  += partials[i * 3 + 2];
  }
#pragma unroll
  for (int o = 16; o > 0; o >>= 1) {
    sR  += __shfl_down(sR,  o, 32);
    sR2 += __shfl_down(sR2, o, 32);
    sA  += __shfl_down(sA,  o, 32);
  }
  if (lane == 0) {
    const float Bf = (float)B_ROWS;
    float return_loss = -sR / Bf;                               // -mean(returns)
    float var = (sR2 - sR * sR / Bf) / (Bf - 1.0f);             // unbiased variance
    float reg = sA / ((float)B_ROWS * (float)N_COLS);           // mean |pred|
    out[0] = 1.0f * return_loss + 0.1f * var + 0.01f * reg;
  }
}

// ---------------- Entry point ----------------
extern "C" void kernel_launch(void* const* d_in, const int* in_sizes, int n_in,
                              void* d_out, int out_size, void* d_ws, size_t ws_size,
                              hipStream_t stream) {
  (void)in_sizes; (void)n_in; (void)out_size; (void)ws_size;
  const float* pred  = (const float*)d_in[0];
  const float* price = (const float*)d_in[1];
  float* out      = (float*)d_out;
  float* partials = (float*)d_ws;          // 64 * 3 floats

  const int nblk = B_ROWS / BROWS;         // 64 blocks
  trade_scan_kernel<<<nblk, BROWS, 0, stream>>>(pred, price, partials);
  finalize_kernel<<<1, 32, 0, stream>>>(partials, out, nblk);
}